// PRME_82111184765586
// MI455X (gfx1250) — compile-verified
//
#include <hip/hip_runtime.h>

// FM forward: B=16384 rows, NNZ=50, K=64, table 1M x 64 f32 (256 MB).
// Memory-bound random gather (~210 MB @ 23.3 TB/s ≈ 9 us ideal).
// Phase 1: wave-per-row coalesced 256B gathers (global_load_b64 per lane).
// Phase 2: per-16-row epilogue sum(emb^2) via chained V_WMMA_F32_16X16X4_F32
//          with all-ones B (layout-proof) and documented A/D layouts.

typedef float v2f __attribute__((ext_vector_type(2)));
typedef float v8f __attribute__((ext_vector_type(8)));

#define FM_NNZ 50
#define FM_K   64
#define ROWS_PER_WG 16
#define E_STRIDE 66   // 64 + 2 pad: float2-aligned, spreads LDS banks

__global__ __launch_bounds__(256)
void PRME_82111184765586_fm_kernel(const int*   __restrict__ feat_idx,
                                   const float* __restrict__ feat_val,
                                   const float* __restrict__ var_linear,
                                   const float* __restrict__ var_factors,
                                   float*       __restrict__ out,
                                   int B)
{
    __shared__ __align__(16) float Elds[ROWS_PER_WG][E_STRIDE]; // emb_mul tile [16][64]
    __shared__ float Dlds[ROWS_PER_WG][17];                     // WMMA D scatter (+pad)
    __shared__ float Plds[ROWS_PER_WG];                         // prod_term
    __shared__ float Llds[ROWS_PER_WG];                         // linear_bias
    __shared__ float Flds[ROWS_PER_WG];                         // feats_sum

    const int lane    = threadIdx.x & 31;
    const int wave    = threadIdx.x >> 5;
    const int rowBase = blockIdx.x * ROWS_PER_WG;

    // ---------------- Phase 1: gather + accumulate (2 rows per wave) ----------
    for (int rr = 0; rr < 2; ++rr) {
        const int lrow = wave * 2 + rr;          // 0..15 within workgroup
        const int row  = rowBase + lrow;
        if (row >= B) break;                     // B % 16 == 0 in practice
        const int base = row * FM_NNZ;

        // preload this row's nnz into lane registers for register broadcast
        int   iA = feat_idx[base + lane];                       // n = lane (0..31 < 50)
        float vA = feat_val[base + lane];
        int   iB = 0; float vB = 0.0f;
        if (lane + 32 < FM_NNZ) {                               // n = lane+32 (32..49)
            iB = feat_idx[base + lane + 32];
            vB = feat_val[base + lane + 32];
        }

        float accx = 0.f, accy = 0.f;   // emb_mul[k], k = 2*lane, 2*lane+1
        float sq   = 0.f;               // prod_term partial

        #pragma unroll
        for (int n = 0; n < FM_NNZ; ++n) {
            int   idx = (n < 32) ? __shfl(iA, n, 32) : __shfl(iB, n - 32, 32);
            float val = (n < 32) ? __shfl(vA, n, 32) : __shfl(vB, n - 32, 32);
            // one 256B factor row per wave: 32 lanes x global_load_b64, coalesced
            const v2f f = *(const v2f*)(var_factors + (size_t)idx * FM_K + 2 * lane);
            accx = fmaf(val, f.x, accx);
            accy = fmaf(val, f.y, accy);
            sq   = fmaf(val, fmaf(f.x, f.x, f.y * f.y), sq);
        }

        // linear term + feats_sum (reuse preloaded idx/val; random dword gather
        // into a 4MB table -> effectively L2-resident)
        float fsum = vA + vB;
        float lb   = vA * var_linear[iA];
        if (lane + 32 < FM_NNZ) lb = fmaf(vB, var_linear[iB], lb);

        // wave32 reductions
        #pragma unroll
        for (int off = 16; off > 0; off >>= 1) {
            sq   += __shfl_xor(sq,   off, 32);
            fsum += __shfl_xor(fsum, off, 32);
            lb   += __shfl_xor(lb,   off, 32);
        }

        // stash: emb_mul row -> LDS (ds_store_b64), scalars from lane 0
        *(v2f*)&Elds[lrow][2 * lane] = (v2f){accx, accy};
        if (lane == 0) { Plds[lrow] = sq; Llds[lrow] = lb; Flds[lrow] = fsum; }
    }

    __syncthreads();

    // ---------------- Phase 2: term2 = sum_k emb^2 via WMMA (wave 0) ----------
    if (wave == 0) {
        // A layout (32-bit 16x4, documented): lane<16 holds (M=lane,K=0|1),
        // lane>=16 holds (M=lane-16,K=2|3). B = all ones -> D[m][*] = rowsum(A[m]).
        const int r  = lane & 15;
        const int hi = lane >> 4;       // selects K-pair {0,1} vs {2,3}
        v8f c = {0.f, 0.f, 0.f, 0.f, 0.f, 0.f, 0.f, 0.f};
        v2f ones; ones.x = 1.0f; ones.y = 1.0f;

        #pragma unroll
        for (int ch = 0; ch < 16; ++ch) {       // 16 chunks of K=4 cover K=64
            const v2f e = *(const v2f*)&Elds[r][ch * 4 + hi * 2];
            v2f a; a.x = e.x * e.x; a.y = e.y * e.y;   // square before row-sum
            c = __builtin_amdgcn_wmma_f32_16x16x4_f32(
                    /*neg_a=*/false, a, /*neg_b=*/false, ones,
                    /*c_mod=*/(short)0, c, /*reuse_a=*/false, /*reuse_b=*/false);
        }

        // D layout (documented): VGPR v, lane<16 -> (M=v, N=lane);
        // lane>=16 -> (M=v+8, N=lane-16). Scatter to LDS, read diagonal.
        #pragma unroll
        for (int v = 0; v < 8; ++v)
            Dlds[v + 8 * hi][r] = c[v];
        // same-wave DS ordering: write->read in order; compiler inserts s_wait_dscnt

        if (lane < ROWS_PER_WG) {
            const int gr = rowBase + lane;
            if (gr < B) {
                const float term2 = Dlds[lane][lane];
                out[gr] = Llds[lane] + Plds[lane] * Flds[lane] - term2;
            }
        }
    }
}

extern "C" void kernel_launch(void* const* d_in, const int* in_sizes, int n_in,
                              void* d_out, int out_size, void* d_ws, size_t ws_size,
                              hipStream_t stream) {
    (void)n_in; (void)d_ws; (void)ws_size;
    const int*   feat_idx    = (const int*)  d_in[0];   // [B, 50] int32
    const float* feat_val    = (const float*)d_in[1];   // [B, 50] f32
    const float* var_linear  = (const float*)d_in[2];   // [1M, 1] f32
    const float* var_factors = (const float*)d_in[3];   // [1M, 64] f32
    float*       out         = (float*)d_out;           // [B, 1] f32

    const int B = in_sizes[0] / FM_NNZ;                 // 16384
    const int grid = (B + ROWS_PER_WG - 1) / ROWS_PER_WG;
    PRME_82111184765586_fm_kernel<<<grid, 256, 0, stream>>>(
        feat_idx, feat_val, var_linear, var_factors, out, B);
    (void)out_size;
}